// AR_79894981640751
// MI455X (gfx1250) — compile-verified
//
#include <hip/hip_runtime.h>

typedef float v2f __attribute__((ext_vector_type(2)));
typedef float v8f __attribute__((ext_vector_type(8)));

#define B_ 8
#define S_ 256
#define D_ 64
#define V_ 8192
#define CHUNK_ 32          // s-slices per mixer block (split-K)
#define NCHUNK_ 8          // S_ / CHUNK_

// ---------------------------------------------------------------------------
// Kernel 1: gather h[b,s,d] = embed[x[b,s], d] into padded [16][S][D] buffer.
// Rows b = 8..15 are zero so the WMMA A-operand (M=16) has clean padding.
// ---------------------------------------------------------------------------
__global__ __launch_bounds__(256) void gather_h(const int* __restrict__ x,
                                                const float* __restrict__ embed,
                                                float* __restrict__ hp) {
    int idx = blockIdx.x * 256 + threadIdx.x;   // 16*256*64 = 262144 total
    int d = idx & 63;
    int s = (idx >> 6) & 255;
    int b = idx >> 14;
    float val = 0.0f;
    if (b < B_) {
        int tok = x[b * S_ + s];
        val = embed[(size_t)tok * D_ + d];
    }
    hp[idx] = val;
}

// ---------------------------------------------------------------------------
// Kernel 2: transpose embed [V][D] -> eT [D][V] (LDS-tiled, coalesced R & W).
// ---------------------------------------------------------------------------
__global__ __launch_bounds__(256) void transpose_embed(const float* __restrict__ embed,
                                                       float* __restrict__ eT) {
    __shared__ float tile[64][65];
    int v0 = blockIdx.x * 64;                    // 128 blocks
    for (int i = threadIdx.x; i < 64 * 64; i += 256) {
        int r = i >> 6, c = i & 63;              // r: v offset, c: o
        tile[r][c] = embed[(size_t)(v0 + r) * D_ + c];
    }
    __syncthreads();
    for (int i = threadIdx.x; i < 64 * 64; i += 256) {
        int r = i >> 6, c = i & 63;              // r: o, c: v offset
        eT[(size_t)r * V_ + v0 + c] = tile[c][r];
    }
}

// ---------------------------------------------------------------------------
// Kernel 3: causal bilinear mix, split-K over s-chunks for load balance.
//   grid = (S_, NCHUNK_); block (t, c) handles s in [c*32, min(c*32+31, t)].
//   y_part[t][c][b][o] += h[b,s,:] x K[t,s,:,:]  via 4 o-tiles x 16 k-steps
//   of v_wmma_f32_16x16x4_f32 per s; cross-wave reduce through LDS.
// Every active block streams <= 512 KB of K -> uniform work, no tail.
// ---------------------------------------------------------------------------
__global__ __launch_bounds__(256) void mixer(const float* __restrict__ Kt,
                                             const float* __restrict__ hp,
                                             float* __restrict__ ypart) {
    const int t     = blockIdx.x;          // 0..255
    const int chunk = blockIdx.y;          // 0..7
    const int s0    = chunk * CHUNK_;
    if (s0 > t) return;                    // block-uniform early exit
    const int send  = (s0 + CHUNK_ - 1 < t) ? (s0 + CHUNK_ - 1) : t;

    const int wave = threadIdx.x >> 5;     // 0..7
    const int lane = threadIdx.x & 31;
    const int half = lane >> 4;            // 0: K rows 0..1, 1: K rows 2..3
    const int l15  = lane & 15;

    v8f acc[4] = {v8f{0.f}, v8f{0.f}, v8f{0.f}, v8f{0.f}};

    for (int s = s0 + wave; s <= send; s += 8) {
        const float* Kts = Kt + ((size_t)t * S_ + s) * D_ * D_;   // [d][o]
        // Preload A fragments for all 16 k-steps (lane: b=l15, d=ks*4+half*2).
        const float* arow = hp + ((size_t)l15 * S_ + s) * D_ + half * 2;
        v2f afrag[16];
#pragma unroll
        for (int ks = 0; ks < 16; ++ks)
            afrag[ks] = *(const v2f*)(arow + ks * 4);

#pragma unroll
        for (int ot = 0; ot < 4; ++ot) {
#pragma unroll
            for (int ks = 0; ks < 16; ++ks) {
                const int d0 = ks * 4 + half * 2;
                v2f bfrag;
                bfrag.x = Kts[(size_t)d0 * D_ + ot * 16 + l15];
                bfrag.y = Kts[(size_t)(d0 + 1) * D_ + ot * 16 + l15];
                acc[ot] = __builtin_amdgcn_wmma_f32_16x16x4_f32(
                    false, afrag[ks], false, bfrag, (short)0, acc[ot],
                    false, false);
            }
        }
    }

    // Deterministic cross-wave reduction through LDS.
    // Acc element (vgpr j, lane L): b = j + 8*(L>=16), o = ot*16 + (L&15).
    // Lanes 0..15 hold the 8 real batch rows.
    __shared__ float red[8][512];   // 16 KB
    if (lane < 16) {
#pragma unroll
        for (int ot = 0; ot < 4; ++ot)
#pragma unroll
            for (int j = 0; j < 8; ++j)
                red[wave][j * 64 + ot * 16 + l15] = acc[ot][j];
    }
    __syncthreads();
    float* yp = ypart + ((size_t)(t * NCHUNK_ + chunk) * B_) * D_;
    for (int i = threadIdx.x; i < 512; i += 256) {
        float sum = 0.0f;
#pragma unroll
        for (int w = 0; w < 8; ++w) sum += red[w][i];
        yp[i] = sum;                       // [b][o] flat, b = i>>6, o = i&63
    }
}

// ---------------------------------------------------------------------------
// Kernel 3b: reduce y_part over chunks -> y[(b*S+t)*64 + o].
// Fixed summation order (c ascending) -> deterministic.
// ---------------------------------------------------------------------------
__global__ __launch_bounds__(256) void reduce_y(const float* __restrict__ ypart,
                                                float* __restrict__ y) {
    int idx = blockIdx.x * 256 + threadIdx.x;   // 16*... use b<8: 8*256*64 = 131072
    int o = idx & 63;
    int t = (idx >> 6) & 255;
    int b = idx >> 14;                           // 0..7
    const int nch = t / CHUNK_ + 1;              // valid chunks for this t
    float sum = 0.0f;
    for (int c = 0; c < nch; ++c)
        sum += ypart[((size_t)(t * NCHUNK_ + c) * B_ + b) * D_ + o];
    y[((size_t)b * S_ + t) * D_ + o] = sum;      // y[m][o], m = b*S + t
}

// ---------------------------------------------------------------------------
// Kernel 4: logits[m,v] = sum_o y[m,o] * eT[o,v].  M=2048, N=8192, K=64.
// Each wave: 16(m) x 64(v) tile = 4 n-tiles x 16 k-steps of WMMA f32.
// ---------------------------------------------------------------------------
__global__ __launch_bounds__(256) void logits_gemm(const float* __restrict__ y,
                                                   const float* __restrict__ eT,
                                                   float* __restrict__ out) {
    const int job  = blockIdx.x * 8 + (threadIdx.x >> 5);  // 16384 wave jobs
    const int lane = threadIdx.x & 31;
    const int half = lane >> 4;
    const int l15  = lane & 15;
    const int mt = job >> 7;          // 0..127
    const int nt = job & 127;         // 0..127 (64 v each)
    const int m0 = mt * 16, v0 = nt * 64;

    v8f acc[4] = {v8f{0.f}, v8f{0.f}, v8f{0.f}, v8f{0.f}};

    const float* arow = y + ((size_t)(m0 + l15)) * D_ + half * 2;
    v2f afrag[16];
#pragma unroll
    for (int ks = 0; ks < 16; ++ks)
        afrag[ks] = *(const v2f*)(arow + ks * 4);

#pragma unroll
    for (int ot = 0; ot < 4; ++ot) {
#pragma unroll
        for (int ks = 0; ks < 16; ++ks) {
            const int k0 = ks * 4 + half * 2;
            v2f bfrag;
            bfrag.x = eT[(size_t)k0 * V_ + v0 + ot * 16 + l15];
            bfrag.y = eT[(size_t)(k0 + 1) * V_ + v0 + ot * 16 + l15];
            acc[ot] = __builtin_amdgcn_wmma_f32_16x16x4_f32(
                false, afrag[ks], false, bfrag, (short)0, acc[ot],
                false, false);
        }
    }

    // Store: (vgpr j, lane L) -> m = m0 + j + 8*half, v = v0 + ot*16 + l15.
#pragma unroll
    for (int ot = 0; ot < 4; ++ot)
#pragma unroll
        for (int j = 0; j < 8; ++j)
            out[(size_t)(m0 + j + 8 * half) * V_ + v0 + ot * 16 + l15] = acc[ot][j];
}

// ---------------------------------------------------------------------------
extern "C" void kernel_launch(void* const* d_in, const int* in_sizes, int n_in,
                              void* d_out, int out_size, void* d_ws, size_t ws_size,
                              hipStream_t stream) {
    const int*   x     = (const int*)d_in[0];     // [B,S]
    const float* embed = (const float*)d_in[1];   // [V,D]
    const float* Kt    = (const float*)d_in[2];   // [S,S,D,D]
    float* out = (float*)d_out;                   // [B,S,V]

    char* ws = (char*)d_ws;
    float* hp    = (float*)(ws);                    // 16*S*D*4      = 1 MB
    float* eT    = (float*)(ws + (1u << 20));       // D*V*4         = 2 MB
    float* y     = (float*)(ws + (3u << 20));       // 2048*64*4     = 512 KB
    float* ypart = (float*)(ws + (4u << 20));       // 256*8*8*64*4  = 4 MB

    gather_h<<<(16 * S_ * D_) / 256, 256, 0, stream>>>(x, embed, hp);
    transpose_embed<<<V_ / 64, 256, 0, stream>>>(embed, eT);
    mixer<<<dim3(S_, NCHUNK_), 256, 0, stream>>>(Kt, hp, ypart);
    reduce_y<<<(B_ * S_ * D_) / 256, 256, 0, stream>>>(ypart, y);
    logits_gemm<<<(128 * 128) / 8, 256, 0, stream>>>(y, eT, out);
}